// PromptBank_39281770889523
// MI455X (gfx1250) — compile-verified
//
#include <hip/hip_runtime.h>
#include <math.h>

typedef float v2f __attribute__((ext_vector_type(2)));
typedef float v8f __attribute__((ext_vector_type(8)));

#define BN   8192     // batch rows (desc)
#define KN   2048     // anchors
#define DN   768      // feature dim
#define MT   64       // M tile per workgroup
#define NT   64       // N tile per workgroup
#define KC   64       // K chunk staged in LDS
#define LST  68       // LDS row stride in dwords (padded, 16B-aligned, conflict-free)
#define NCH  (DN / KC)
#define EPSN 1e-12f
#define MOM  0.9f

// ---------------------------------------------------------------------------
// Kernel 0: inverse L2 norms of desc rows and anchor rows (one wave per row)
// ---------------------------------------------------------------------------
__global__ __launch_bounds__(256) void pb_norms(const float* __restrict__ desc,
                                                const float* __restrict__ anch,
                                                float* __restrict__ invd,
                                                float* __restrict__ inva) {
  const int lane = threadIdx.x & 31;
  const int wave = blockIdx.x * 8 + (threadIdx.x >> 5);
  const float* row;
  float* dst;
  if (wave < BN) {
    row = desc + (size_t)wave * DN;
    dst = invd + wave;
  } else {
    int a = wave - BN;
    if (a >= KN) return;
    row = anch + (size_t)a * DN;
    dst = inva + a;
  }
  float ss = 0.0f;
  #pragma unroll
  for (int e = 0; e < DN / 32; ++e) {
    float x = row[e * 32 + lane];
    ss += x * x;
  }
  #pragma unroll
  for (int m = 1; m < 32; m <<= 1) ss += __shfl_xor(ss, m, 32);
  if (lane == 0) *dst = 1.0f / fmaxf(sqrtf(ss), EPSN);
}

// ---------------------------------------------------------------------------
// Async staging helpers: per-lane global -> LDS copy bypassing VGPRs (ASYNCcnt)
// ---------------------------------------------------------------------------
__device__ __forceinline__ void async_b128(const float* g, const float* l) {
  unsigned lds = (unsigned)(uintptr_t)l;   // low 32 bits = LDS byte offset
  asm volatile("global_load_async_to_lds_b128 %0, %1, off"
               :: "v"(lds), "v"(g) : "memory");
}
#define ASYNC_WAIT_PREV() asm volatile("s_wait_asynccnt 0x8" ::: "memory")
#define ASYNC_WAIT_ALL()  asm volatile("s_wait_asynccnt 0x0" ::: "memory")

// ---------------------------------------------------------------------------
// Kernel 1: S = desc @ anchors^T via V_WMMA_F32_16X16X4_F32, fused row-argmax
// (cosine argmin). Norm scaling folded into the epilogue.
// Workgroup: 8 waves; wave grid 4(M) x 2(N); each wave -> 16x32 outputs.
// Double-buffered LDS tiles filled with GLOBAL_LOAD_ASYNC_TO_LDS_B128 so the
// chunk-(i+1) copy overlaps the chunk-(i) WMMA burst.
// ---------------------------------------------------------------------------
__global__ __launch_bounds__(256) void pb_assign(const float* __restrict__ desc,
                                                 const float* __restrict__ anch,
                                                 const float* __restrict__ invd,
                                                 const float* __restrict__ inva,
                                                 float* __restrict__ ids) {
  __shared__ __align__(16) float lsA[2][MT * LST];
  __shared__ __align__(16) float lsB[2][NT * LST];

  const int tid   = threadIdx.x;
  const int lane  = tid & 31;
  const int wv    = tid >> 5;   // 0..7
  const int wm    = wv & 3;     // M sub-block (16 rows)
  const int wn    = wv >> 2;    // N half (32 cols)
  const int r     = lane & 15;
  const int half  = lane >> 4;
  const int mbase = blockIdx.x * MT;

  // this thread's staging slot: 4 float4s each for A and B per chunk
  const int srow = tid >> 2;          // 0..63 (unused; staging row is idx >> 4)
  (void)srow;

  // per-lane inverse-norm for the 8 M rows this lane's C fragment covers
  float iva[8];
  #pragma unroll
  for (int j = 0; j < 8; ++j) iva[j] = invd[mbase + wm * 16 + half * 8 + j];

  float bv[8];
  int   bi[8];
  #pragma unroll
  for (int j = 0; j < 8; ++j) { bv[j] = -1e30f; bi[j] = 0; }

  for (int nt = 0; nt < KN / NT; ++nt) {
    const int nbase = nt * NT;
    v8f acc0 = {0.f, 0.f, 0.f, 0.f, 0.f, 0.f, 0.f, 0.f};
    v8f acc1 = {0.f, 0.f, 0.f, 0.f, 0.f, 0.f, 0.f, 0.f};

    // prologue: launch async copy of chunk 0 into buffer 0
    #pragma unroll
    for (int t = 0; t < 4; ++t) {
      int idx = t * 256 + tid;
      int row = idx >> 4;
      int c4  = (idx & 15) << 2;
      async_b128(desc + (size_t)(mbase + row) * DN + c4, &lsA[0][row * LST + c4]);
      async_b128(anch + (size_t)(nbase + row) * DN + c4, &lsB[0][row * LST + c4]);
    }

    for (int ci = 0; ci < NCH; ++ci) {
      const int cur = ci & 1;
      if (ci + 1 < NCH) {
        // launch async copy of next chunk into the other buffer, then wait
        // until only those 8 copies remain outstanding (in-order completion
        // => current chunk's LDS writes from this wave are done)
        const int kcn = (ci + 1) * KC;
        const int nxt = cur ^ 1;
        #pragma unroll
        for (int t = 0; t < 4; ++t) {
          int idx = t * 256 + tid;
          int row = idx >> 4;
          int c4  = (idx & 15) << 2;
          async_b128(desc + (size_t)(mbase + row) * DN + kcn + c4,
                     &lsA[nxt][row * LST + c4]);
          async_b128(anch + (size_t)(nbase + row) * DN + kcn + c4,
                     &lsB[nxt][row * LST + c4]);
        }
        ASYNC_WAIT_PREV();
      } else {
        ASYNC_WAIT_ALL();
      }
      __syncthreads();   // current chunk visible to all waves

      // fragment bases: lane holds 2 consecutive K for one row/col
      const float* Ab  = &lsA[cur][(wm * 16 + r) * LST + 2 * half];
      const float* Bb0 = &lsB[cur][(wn * 32 + r) * LST + 2 * half];
      const float* Bb1 = Bb0 + 16 * LST;

      // one-step software pipeline: fetch kk+4 fragments before using kk
      v2f a  = *(const v2f*)(Ab);
      v2f b0 = *(const v2f*)(Bb0);
      v2f b1 = *(const v2f*)(Bb1);
      #pragma unroll
      for (int kk = 0; kk < KC; kk += 4) {
        v2f an = a, b0n = b0, b1n = b1;
        if (kk + 4 < KC) {
          an  = *(const v2f*)(Ab  + kk + 4);
          b0n = *(const v2f*)(Bb0 + kk + 4);
          b1n = *(const v2f*)(Bb1 + kk + 4);
        }
        acc0 = __builtin_amdgcn_wmma_f32_16x16x4_f32(false, a, false, b0,
                                                     (short)0, acc0, false, false);
        acc1 = __builtin_amdgcn_wmma_f32_16x16x4_f32(false, a, false, b1,
                                                     (short)0, acc1, false, false);
        a = an; b0 = b0n; b1 = b1n;
      }
      __syncthreads();   // all waves done reading buf[cur] before it's refilled
    }

    // epilogue: scale by inverse norms, running argmax (first-min tie rule)
    const int   c0   = nbase + wn * 32 + r;
    const int   c1   = c0 + 16;
    const float ivb0 = inva[c0];
    const float ivb1 = inva[c1];
    #pragma unroll
    for (int j = 0; j < 8; ++j) {
      float v0 = acc0[j] * iva[j] * ivb0;
      if (v0 > bv[j] || (v0 == bv[j] && c0 < bi[j])) { bv[j] = v0; bi[j] = c0; }
      float v1 = acc1[j] * iva[j] * ivb1;
      if (v1 > bv[j] || (v1 == bv[j] && c1 < bi[j])) { bv[j] = v1; bi[j] = c1; }
    }
  }

  // cross-lane argmax butterfly within each 16-lane half
  #pragma unroll
  for (int m = 1; m < 16; m <<= 1) {
    #pragma unroll
    for (int j = 0; j < 8; ++j) {
      float ov = __shfl_xor(bv[j], m, 32);
      int   oi = __shfl_xor(bi[j], m, 32);
      if (ov > bv[j] || (ov == bv[j] && oi < bi[j])) { bv[j] = ov; bi[j] = oi; }
    }
  }
  if (r == 0) {
    #pragma unroll
    for (int j = 0; j < 8; ++j)
      ids[mbase + wm * 16 + half * 8 + j] = (float)bi[j];
  }
}

// ---------------------------------------------------------------------------
// Kernel 2: exact sequential EMA semantics, parallelized per-anchor.
// One wave per anchor; scans ids[] in batch order with __ballot so each
// anchor's update chain is applied in the same order as the reference scan.
// ---------------------------------------------------------------------------
__global__ __launch_bounds__(256) void pb_ema(const float* __restrict__ desc,
                                              const float* __restrict__ anch,
                                              const float* __restrict__ cnts_in,
                                              const float* __restrict__ ids,
                                              float* __restrict__ anch_out,
                                              float* __restrict__ cnts_out) {
  const int lane = threadIdx.x & 31;
  const int k    = blockIdx.x * 8 + (threadIdx.x >> 5);
  if (k >= KN) return;

  float a[DN / 32];
  #pragma unroll
  for (int e = 0; e < DN / 32; ++e) a[e] = anch[(size_t)k * DN + e * 32 + lane];
  float cnt = cnts_in[k];

  for (int b0 = 0; b0 < BN; b0 += 32) {
    int id = (int)ids[b0 + lane];
    unsigned mask = (unsigned)__ballot(id == k);
    while (mask) {
      int b = b0 + (__ffs(mask) - 1);
      mask &= mask - 1;
      float m  = (cnt < 1.0f) ? 0.0f : MOM;
      float om = 1.0f - m;
      float ss = 0.0f;
      #pragma unroll
      for (int e = 0; e < DN / 32; ++e) {
        float d = desc[(size_t)b * DN + e * 32 + lane];  // raw desc, as in ref
        a[e] = m * a[e] + om * d;
        ss += a[e] * a[e];
      }
      #pragma unroll
      for (int mm = 1; mm < 32; mm <<= 1) ss += __shfl_xor(ss, mm, 32);
      float inv = 1.0f / fmaxf(sqrtf(ss), EPSN);
      #pragma unroll
      for (int e = 0; e < DN / 32; ++e) a[e] *= inv;
      cnt += 1.0f;
    }
  }

  #pragma unroll
  for (int e = 0; e < DN / 32; ++e)
    anch_out[(size_t)k * DN + e * 32 + lane] = a[e];
  if (lane == 0) cnts_out[k] = cnt;
}

// ---------------------------------------------------------------------------
extern "C" void kernel_launch(void* const* d_in, const int* in_sizes, int n_in,
                              void* d_out, int out_size, void* d_ws, size_t ws_size,
                              hipStream_t stream) {
  (void)in_sizes; (void)n_in; (void)out_size; (void)d_ws; (void)ws_size;
  const float* desc = (const float*)d_in[0];  // [8192,768]
  const float* anch = (const float*)d_in[1];  // [2048,768]
  const float* cnts = (const float*)d_in[2];  // [2048]

  float* out       = (float*)d_out;
  float* ids       = out;                           // [8192]
  float* anch_out  = out + BN;                      // [2048*768]
  float* cnts_out  = out + BN + (size_t)KN * DN;    // [2048]

  // scratch inside the (not-yet-written) anchors output region
  float* invd = anch_out;        // [8192] inverse norms of desc rows
  float* inva = anch_out + BN;   // [2048] inverse norms of anchor rows

  pb_norms<<<(BN + KN) / 8, 256, 0, stream>>>(desc, anch, invd, inva);
  pb_assign<<<BN / MT, 256, 0, stream>>>(desc, anch, invd, inva, ids);
  pb_ema<<<KN / 8, 256, 0, stream>>>(desc, anch, cnts, ids, anch_out, cnts_out);
}